// DeepLoRACollectiveIntelligence_48232482734585
// MI455X (gfx1250) — compile-verified
//
#include <hip/hip_runtime.h>
#include <hip/hip_bf16.h>
#include <math.h>

typedef __attribute__((ext_vector_type(2))) float v2f;
typedef __attribute__((ext_vector_type(8))) float v8f;

#define NEGF (-3.0e38f)
#define JW 4  // waves per block splitting the j (column) range

static __device__ __forceinline__ v8f wmma4(v2f a, v2f b, v8f c) {
  // D = A(16x4 f32) x B(4x16 f32) + C(16x16 f32)
  return __builtin_amdgcn_wmma_f32_16x16x4_f32(false, a, false, b, (short)0, c, false, false);
}

static __device__ __forceinline__ float wave_reduce_sum16(float v) {
  for (int off = 1; off < 16; off <<= 1) v += __shfl_xor(v, off, 32);
  return v;
}
static __device__ __forceinline__ float wave_reduce_max16(float v) {
  for (int off = 1; off < 16; off <<= 1) v = fmaxf(v, __shfl_xor(v, off, 32));
  return v;
}
static __device__ __forceinline__ float wave_reduce_sum32(float v) {
  for (int off = 1; off < 32; off <<= 1) v += __shfl_xor(v, off, 32);
  return v;
}

// ---------------------------------------------------------------------------
// Generic f32 WMMA GEMM: C[M,N] = A[M,K] @ B[K,N] (+ bias[n]); M,N %16==0, K %4==0
// One wave per 16x16 output tile.
// ---------------------------------------------------------------------------
__global__ void gemm_wmma_f32(const float* __restrict__ A, const float* __restrict__ B,
                              const float* __restrict__ bias, float* __restrict__ C,
                              int M, int N, int K) {
  int wave = (int)((blockIdx.x * blockDim.x + threadIdx.x) >> 5);
  int lane = (int)(threadIdx.x & 31);
  int tilesN = N >> 4;
  int tilesM = M >> 4;
  int tiM = wave / tilesN;
  int tiN = wave - tiM * tilesN;
  if (tiM >= tilesM) return;  // wave-uniform
  int m0 = tiM << 4, n0 = tiN << 4;
  int hi = lane >> 4, col = lane & 15;

  v8f acc = {};
  const float* __restrict__ Arow = A + (size_t)(m0 + col) * K + 2 * hi;
  const float* __restrict__ Bc   = B + n0 + col;
#pragma unroll 4
  for (int k0 = 0; k0 < K; k0 += 4) {
    v2f a; a.x = Arow[k0]; a.y = Arow[k0 + 1];
    v2f b;
    b.x = Bc[(size_t)(k0 + 2 * hi) * N];
    b.y = Bc[(size_t)(k0 + 2 * hi + 1) * N];
    acc = wmma4(a, b, acc);
  }
  float bv = bias ? bias[n0 + col] : 0.0f;
#pragma unroll
  for (int r = 0; r < 8; ++r) {
    C[(size_t)(m0 + r + 8 * hi) * N + n0 + col] = acc[r] + bv;
  }
}

// ---------------------------------------------------------------------------
// Row-wise LayerNorm (+ optional ReLU), in-place. One wave per row.
// ---------------------------------------------------------------------------
__global__ void ln_rows(float* __restrict__ X, const float* __restrict__ g,
                        const float* __restrict__ b, int W, int relu) {
  int row = (int)blockIdx.x;
  int lane = (int)threadIdx.x;
  float* x = X + (size_t)row * W;
  float s = 0.f, s2 = 0.f;
  for (int i = lane; i < W; i += 32) { float v = x[i]; s += v; s2 += v * v; }
  s = wave_reduce_sum32(s);
  s2 = wave_reduce_sum32(s2);
  float mean = s / (float)W;
  float var = s2 / (float)W - mean * mean;
  float inv = rsqrtf(var + 1e-5f);
  for (int i = lane; i < W; i += 32) {
    float v = (x[i] - mean) * inv * g[i] + b[i];
    if (relu) v = fmaxf(v, 0.0f);
    x[i] = v;
  }
}

// ---------------------------------------------------------------------------
// ti[i] = dot(tanh(temperaments[i] @ temp_W + temp_b), gate_wa).  Wave per row.
// ---------------------------------------------------------------------------
__global__ void ti_kernel(const float* __restrict__ temps, const float* __restrict__ tW,
                          const float* __restrict__ tb, const float* __restrict__ gwa,
                          float* __restrict__ ti) {
  int row = (int)blockIdx.x;
  int lane = (int)threadIdx.x;
  float acc = 0.f;
  for (int e = lane; e < 128; e += 32) {
    float t = tb[e];
#pragma unroll
    for (int tt = 0; tt < 10; ++tt) t += temps[(size_t)row * 10 + tt] * tW[tt * 128 + e];
    acc += tanhf(t) * gwa[e];
  }
  acc = wave_reduce_sum32(acc);
  if (lane == 0) ti[row] = acc;
}

// cj[i] = dot(ctx[i], gate_wb). Wave per row.
__global__ void cj_kernel(const float* __restrict__ ctx, const float* __restrict__ gwb,
                          float* __restrict__ cj) {
  int row = (int)blockIdx.x;
  int lane = (int)threadIdx.x;
  float acc = 0.f;
  for (int e = lane; e < 128; e += 32) acc += ctx[(size_t)row * 128 + e] * gwb[e];
  acc = wave_reduce_sum32(acc);
  if (lane == 0) cj[row] = acc;
}

// cnt[i] = sum_j adj[i,j] - adj[i,i]   (neighbors excluding self)
__global__ void cnt_kernel(const int* __restrict__ adj, float* __restrict__ cnt, int N) {
  int row = (int)blockIdx.x;
  int lane = (int)threadIdx.x;
  float s = 0.f;
  const int* ar = adj + (size_t)row * N;
  for (int j = lane; j < N; j += 32) s += (ar[j] != 0) ? 1.0f : 0.0f;
  s = wave_reduce_sum32(s);
  if (lane == 0) cnt[row] = s - ((ar[row] != 0) ? 1.0f : 0.0f);
}

// ---------------------------------------------------------------------------
// Flash-style adjacency-masked MHA, j-range split across JW waves per block.
// Block = (head, 16-row block); wave w covers columns [w*N/JW, (w+1)*N/JW).
// Partial (m, l, ctx) merged by wave 0 via log-sum-exp combine.
// ---------------------------------------------------------------------------
__global__ void __launch_bounds__(32 * JW)
flash_attn(const float* __restrict__ q, const float* __restrict__ k,
           const float* __restrict__ v, const int* __restrict__ adj,
           float* __restrict__ ctxh, int N) {
  const int E = 128, DH = 32;
  int wid = (int)blockIdx.x;
  int h = wid & 3;
  int i0 = (wid >> 2) << 4;
  int wv = (int)(threadIdx.x >> 5);
  int lane = (int)(threadIdx.x & 31);
  int hi = lane >> 4, col = lane & 15;
  const float scale = 0.17677669529663688f;  // 1/sqrt(32)

  __shared__ float Kt[JW][16 * 33];
  __shared__ float Vt[JW][16 * 33];
  __shared__ float Pt[JW][16 * 17];
  __shared__ float Cw[JW][16][32];
  __shared__ float Mw[JW][16];
  __shared__ float Lw[JW][16];

  // preload scaled Q fragments (A layout, K=32 -> 8 steps)
  v2f qr[8];
  const float* Qrow = q + (size_t)(i0 + col) * E + h * DH + 2 * hi;
#pragma unroll
  for (int s = 0; s < 8; ++s) { qr[s].x = Qrow[4 * s] * scale; qr[s].y = Qrow[4 * s + 1] * scale; }

  float mrun[8], lrun[8];
#pragma unroll
  for (int r = 0; r < 8; ++r) { mrun[r] = NEGF; lrun[r] = 0.f; }
  v8f acc0 = {}, acc1 = {};

  const int jspan = N / JW;
  const int jbeg = wv * jspan, jend = jbeg + jspan;
  float* Ktw = Kt[wv];
  float* Vtw = Vt[wv];
  float* Ptw = Pt[wv];

  for (int j0 = jbeg; j0 < jend; j0 += 16) {
    // cooperative K/V tile load (16 x 32 each), padded stride 33
    for (int t = lane; t < 128; t += 32) {
      int rr = t >> 3, cc = (t & 7) * 4;
      const float4 kv = *(const float4*)(k + (size_t)(j0 + rr) * E + h * DH + cc);
      const float4 vv = *(const float4*)(v + (size_t)(j0 + rr) * E + h * DH + cc);
      if (j0 + 16 < jend) {  // one-tile lookahead prefetch (global_prefetch_b8)
        __builtin_prefetch(k + (size_t)(j0 + 16 + rr) * E + h * DH + cc, 0, 0);
        __builtin_prefetch(v + (size_t)(j0 + 16 + rr) * E + h * DH + cc, 0, 0);
      }
      float* kp = &Ktw[rr * 33 + cc];
      kp[0] = kv.x; kp[1] = kv.y; kp[2] = kv.z; kp[3] = kv.w;
      float* vp = &Vtw[rr * 33 + cc];
      vp[0] = vv.x; vp[1] = vv.y; vp[2] = vv.z; vp[3] = vv.w;
    }
    // S = Qs @ K^T   (B[kk][n] = K[j0+n][kk] from LDS)
    v8f s = {};
#pragma unroll
    for (int k0 = 0; k0 < 32; k0 += 4) {
      v2f a = qr[k0 >> 2];
      v2f b;
      b.x = Ktw[col * 33 + k0 + 2 * hi];
      b.y = Ktw[col * 33 + k0 + 2 * hi + 1];
      s = wmma4(a, b, s);
    }
    // mask + online softmax per row
#pragma unroll
    for (int r = 0; r < 8; ++r) {
      int grow = i0 + r + 8 * hi;
      int gcol = j0 + col;
      int av = adj[(size_t)grow * N + gcol];
      if (j0 + 16 < jend)
        __builtin_prefetch(&adj[(size_t)grow * N + gcol + 16], 0, 0);
      bool ok = (av != 0) || (grow == gcol);
      float sv = ok ? s[r] : NEGF;
      float tmax = wave_reduce_max16(sv);
      float mnew = fmaxf(mrun[r], tmax);
      float al = __expf(mrun[r] - mnew);
      float p = ok ? __expf(s[r] - mnew) : 0.0f;
      float psum = wave_reduce_sum16(p);
      lrun[r] = lrun[r] * al + psum;
      mrun[r] = mnew;
      acc0[r] *= al;
      acc1[r] *= al;
      Ptw[(r + 8 * hi) * 17 + col] = p;
    }
    // acc += P @ Vt   (P round-tripped through LDS into A layout)
#pragma unroll
    for (int k0 = 0; k0 < 16; k0 += 4) {
      v2f a;
      a.x = Ptw[col * 17 + k0 + 2 * hi];
      a.y = Ptw[col * 17 + k0 + 2 * hi + 1];
      v2f b0, b1;
      b0.x = Vtw[(k0 + 2 * hi) * 33 + col];
      b0.y = Vtw[(k0 + 2 * hi + 1) * 33 + col];
      acc0 = wmma4(a, b0, acc0);
      b1.x = Vtw[(k0 + 2 * hi) * 33 + 16 + col];
      b1.y = Vtw[(k0 + 2 * hi + 1) * 33 + 16 + col];
      acc1 = wmma4(a, b1, acc1);
    }
  }

  // publish partials
  if (col == 0) {
#pragma unroll
    for (int r = 0; r < 8; ++r) { Mw[wv][r + 8 * hi] = mrun[r]; Lw[wv][r + 8 * hi] = lrun[r]; }
  }
#pragma unroll
  for (int r = 0; r < 8; ++r) {
    Cw[wv][r + 8 * hi][col] = acc0[r];
    Cw[wv][r + 8 * hi][16 + col] = acc1[r];
  }
  __syncthreads();

  // wave 0: log-sum-exp combine of JW partials, write ctx
  if (wv == 0) {
#pragma unroll
    for (int r = 0; r < 8; ++r) {
      int row = r + 8 * hi;
      float m = Mw[0][row];
#pragma unroll
      for (int w = 1; w < JW; ++w) m = fmaxf(m, Mw[w][row]);
      float l = 0.f, v0 = 0.f, v1 = 0.f;
#pragma unroll
      for (int w = 0; w < JW; ++w) {
        float e = __expf(Mw[w][row] - m);
        l += Lw[w][row] * e;
        v0 += Cw[w][row][col] * e;
        v1 += Cw[w][row][16 + col] * e;
      }
      float invl = 1.0f / l;  // diagonal guarantees l > 0
      ctxh[(size_t)(i0 + row) * E + h * DH + col] = v0 * invl;
      ctxh[(size_t)(i0 + row) * E + h * DH + 16 + col] = v1 * invl;
    }
  }
}

// ---------------------------------------------------------------------------
// Personality-gated neighbor aggregation, j-range split across JW waves.
// Block = (16-row block, 32-col quarter). Partials are plain sums.
// updated = emb + 0.2 * [(nmask*gate) @ ctx] / max(cnt,1)   (0 if cnt==0)
// ---------------------------------------------------------------------------
__global__ void __launch_bounds__(32 * JW)
gate_agg(const float* __restrict__ ctx, const float* __restrict__ emb,
         const float* __restrict__ ti, const float* __restrict__ cj,
         const float* __restrict__ gate_b_p, const int* __restrict__ adj,
         const float* __restrict__ cnt, float* __restrict__ updated, int N) {
  const int E = 128;
  int wid = (int)blockIdx.x;
  int np = wid & 3;
  int i0 = (wid >> 2) << 4;
  int c0 = np * 32;
  int wv = (int)(threadIdx.x >> 5);
  int lane = (int)(threadIdx.x & 31);
  int hi = lane >> 4, col = lane & 15;

  __shared__ float Gt[JW][16 * 17];
  __shared__ float Cw[JW][16][32];

  float gb = gate_b_p[0];
  float tiv[8];
#pragma unroll
  for (int r = 0; r < 8; ++r) tiv[r] = ti[i0 + r + 8 * hi];

  v8f acc0 = {}, acc1 = {};
  const int jspan = N / JW;
  const int jbeg = wv * jspan, jend = jbeg + jspan;
  float* Gtw = Gt[wv];

  for (int j0 = jbeg; j0 < jend; j0 += 16) {
    float cjv = cj[j0 + col];
#pragma unroll
    for (int r = 0; r < 8; ++r) {
      int grow = i0 + r + 8 * hi;
      int gcol = j0 + col;
      int av = adj[(size_t)grow * N + gcol];
      if (j0 + 16 < jend)
        __builtin_prefetch(&adj[(size_t)grow * N + gcol + 16], 0, 0);
      float g = 0.0f;
      if (av != 0 && grow != gcol)
        g = 1.0f / (1.0f + __expf(-(tiv[r] + cjv + gb)));
      Gtw[(r + 8 * hi) * 17 + col] = g;
    }
#pragma unroll
    for (int k0 = 0; k0 < 16; k0 += 4) {
      v2f a;
      a.x = Gtw[col * 17 + k0 + 2 * hi];
      a.y = Gtw[col * 17 + k0 + 2 * hi + 1];
      const float* B0 = ctx + (size_t)(j0 + k0 + 2 * hi) * E + c0;
      const float* B1 = ctx + (size_t)(j0 + k0 + 2 * hi + 1) * E + c0;
      v2f b0; b0.x = B0[col]; b0.y = B1[col];
      acc0 = wmma4(a, b0, acc0);
      v2f b1; b1.x = B0[16 + col]; b1.y = B1[16 + col];
      acc1 = wmma4(a, b1, acc1);
    }
  }

#pragma unroll
  for (int r = 0; r < 8; ++r) {
    Cw[wv][r + 8 * hi][col] = acc0[r];
    Cw[wv][r + 8 * hi][16 + col] = acc1[r];
  }
  __syncthreads();

  if (wv == 0) {
#pragma unroll
    for (int r = 0; r < 8; ++r) {
      int row = r + 8 * hi;
      int grow = i0 + row;
      float s0 = 0.f, s1 = 0.f;
#pragma unroll
      for (int w = 0; w < JW; ++w) { s0 += Cw[w][row][col]; s1 += Cw[w][row][16 + col]; }
      float c = cnt[grow];
      float invc = (c > 0.0f) ? 0.2f / fmaxf(c, 1.0f) : 0.0f;
      size_t base = (size_t)grow * E + c0;
      updated[base + col] = emb[base + col] + s0 * invc;
      updated[base + 16 + col] = emb[base + 16 + col] + s1 * invc;
    }
  }
}

// ---------------------------------------------------------------------------
// Generational memory retrieval (G=10): updated = 0.9*u + 0.1*softmax(u@mb^T/sqrt(E))@mb
// One wave per row, in-place.
// ---------------------------------------------------------------------------
__global__ void memory_kernel(float* __restrict__ updated, const float* __restrict__ mb) {
  const int E = 128, G = 10;
  int row = (int)blockIdx.x;
  int lane = (int)threadIdx.x;
  float u[4];
#pragma unroll
  for (int i = 0; i < 4; ++i) u[i] = updated[(size_t)row * E + lane + i * 32];
  float sc[G];
#pragma unroll
  for (int g = 0; g < G; ++g) {
    float d = 0.f;
#pragma unroll
    for (int i = 0; i < 4; ++i) d += u[i] * mb[g * E + lane + i * 32];
    d = wave_reduce_sum32(d);
    sc[g] = d * 0.08838834764831845f;  // 1/sqrt(128)
  }
  float mx = sc[0];
#pragma unroll
  for (int g = 1; g < G; ++g) mx = fmaxf(mx, sc[g]);
  float se = 0.f;
#pragma unroll
  for (int g = 0; g < G; ++g) { sc[g] = __expf(sc[g] - mx); se += sc[g]; }
  float inv = 1.0f / se;
#pragma unroll
  for (int i = 0; i < 4; ++i) {
    float anc = 0.f;
#pragma unroll
    for (int g = 0; g < G; ++g) anc += sc[g] * mb[g * E + lane + i * 32];
    updated[(size_t)row * E + lane + i * 32] = 0.9f * u[i] + 0.1f * anc * inv;
  }
}

// ---------------------------------------------------------------------------
static inline void launch_gemm(const float* A, const float* B, const float* bias, float* C,
                               int M, int N, int K, hipStream_t s) {
  int waves = (M / 16) * (N / 16);
  int wpb = 8;  // 256 threads
  int blocks = (waves + wpb - 1) / wpb;
  gemm_wmma_f32<<<blocks, 256, 0, s>>>(A, B, bias, C, M, N, K);
}

extern "C" void kernel_launch(void* const* d_in, const int* in_sizes, int n_in,
                              void* d_out, int out_size, void* d_ws, size_t ws_size,
                              hipStream_t stream) {
  const int N = 4096, E = 128;
  const float* lora  = (const float*)d_in[0];
  const float* temps = (const float*)d_in[1];
  const int*   adj   = (const int*)d_in[2];
  const float* mb    = (const float*)d_in[3];
  const float* eW1 = (const float*)d_in[4];  const float* eb1 = (const float*)d_in[5];
  const float* eg1 = (const float*)d_in[6];  const float* ebe1 = (const float*)d_in[7];
  const float* eW2 = (const float*)d_in[8];  const float* eb2 = (const float*)d_in[9];
  const float* eg2 = (const float*)d_in[10]; const float* ebe2 = (const float*)d_in[11];
  const float* tW  = (const float*)d_in[12]; const float* tb  = (const float*)d_in[13];
  const float* Wq  = (const float*)d_in[14]; const float* Wk  = (const float*)d_in[15];
  const float* Wv  = (const float*)d_in[16]; const float* Wo  = (const float*)d_in[17];
  const float* gwa = (const float*)d_in[18]; const float* gwb = (const float*)d_in[19];
  const float* gbp = (const float*)d_in[20];
  const float* dW1 = (const float*)d_in[21]; const float* db1 = (const float*)d_in[22];
  const float* dg1 = (const float*)d_in[23]; const float* dbe1 = (const float*)d_in[24];
  const float* dW2 = (const float*)d_in[25]; const float* db2 = (const float*)d_in[26];
  float* out = (float*)d_out;

  // workspace allocator (floats), ~23 MB total
  float* f = (float*)d_ws;
  float* buf512 = f;                 f += (size_t)N * 512;   // encoder/decoder hidden
  float* emb    = f;                 f += (size_t)N * E;
  float* qb     = f;                 f += (size_t)N * E;
  float* kb     = f;                 f += (size_t)N * E;
  float* vb     = f;                 f += (size_t)N * E;
  float* ctxh   = f;                 f += (size_t)N * E;
  float* ctx    = f;                 f += (size_t)N * E;
  float* upd    = f;                 f += (size_t)N * E;
  float* tiv    = f;                 f += N;
  float* cjv    = f;                 f += N;
  float* cntv   = f;                 f += N;

  // --- encoder ---
  launch_gemm(lora, eW1, eb1, buf512, N, 512, 256, stream);
  ln_rows<<<N, 32, 0, stream>>>(buf512, eg1, ebe1, 512, 1);
  launch_gemm(buf512, eW2, eb2, emb, N, E, 512, stream);
  ln_rows<<<N, 32, 0, stream>>>(emb, eg2, ebe2, E, 0);

  // --- Q/K/V projections ---
  launch_gemm(emb, Wq, nullptr, qb, N, E, E, stream);
  launch_gemm(emb, Wk, nullptr, kb, N, E, E, stream);
  launch_gemm(emb, Wv, nullptr, vb, N, E, E, stream);

  // --- masked multi-head attention (flash, split-j) + output projection ---
  flash_attn<<<4 * (N / 16), 32 * JW, 0, stream>>>(qb, kb, vb, adj, ctxh, N);
  launch_gemm(ctxh, Wo, nullptr, ctx, N, E, E, stream);

  // --- gate scalars, degree counts ---
  ti_kernel<<<N, 32, 0, stream>>>(temps, tW, tb, gwa, tiv);
  cj_kernel<<<N, 32, 0, stream>>>(ctx, gwb, cjv);
  cnt_kernel<<<N, 32, 0, stream>>>(adj, cntv, N);

  // --- gated neighbor aggregation (split-j) ---
  gate_agg<<<4 * (N / 16), 32 * JW, 0, stream>>>(ctx, emb, tiv, cjv, gbp, adj, cntv, upd, N);

  // --- generational memory retrieval (in-place on upd) ---
  memory_kernel<<<N, 32, 0, stream>>>(upd, mb);

  // --- decoder ---
  launch_gemm(upd, dW1, db1, buf512, N, 512, E, stream);
  ln_rows<<<N, 32, 0, stream>>>(buf512, dg1, dbe1, 512, 1);
  launch_gemm(buf512, dW2, db2, out, N, 256, 512, stream);

  (void)in_sizes; (void)n_in; (void)out_size; (void)ws_size;
}